// LambdaAttention_20718922236172
// MI455X (gfx1250) — compile-verified
//
#include <hip/hip_runtime.h>
#include <cmath>

typedef __attribute__((ext_vector_type(2))) float v2f;
typedef __attribute__((ext_vector_type(8))) float v8f;

#define WMMA_F32X4(A, B, C) \
  __builtin_amdgcn_wmma_f32_16x16x4_f32(false, (A), false, (B), (short)0, (C), false, false)

// Low 32 bits of a generic shared-memory pointer are the LDS byte offset
// (ISA 10.2 aperture mapping).
__device__ __forceinline__ unsigned lds_byte_off(const void* p) {
  return (unsigned)(uintptr_t)p;
}

// Async copy 16B from global to LDS (gfx1250 GLOBAL_LOAD_ASYNC_TO_LDS_B128,
// tracked by ASYNCcnt). VDST = per-lane LDS byte address, VADDR = 64-bit VA.
__device__ __forceinline__ void async_g2lds_b128(unsigned lds_addr, const void* gaddr) {
  asm volatile("global_load_async_to_lds_b128 %0, %1, off"
               :: "v"(lds_addr), "v"((unsigned long long)(uintptr_t)gaddr)
               : "memory");
}

__device__ __forceinline__ void wait_async_zero() {
  asm volatile("s_wait_asynccnt 0x0" ::: "memory");
}

// ---------------------------------------------------------------------------
// Kernel 1: fused q/k/v projection GEMM + BN epilogue.
// out rows 0..63  -> q (BN), 64..79 -> keys (raw), 80..143 -> values (BN)
// grid: (32 col-blocks, 9 row-blocks, 8 batches), block: 256 (8 waves)
// ---------------------------------------------------------------------------
__global__ __launch_bounds__(256) void proj_kernel(
    const float* __restrict__ x,                          // [8,256,4096]
    const float* __restrict__ Wq, const float* __restrict__ qg,
    const float* __restrict__ qb, const float* __restrict__ qm,
    const float* __restrict__ qv,
    const float* __restrict__ Wk, const float* __restrict__ Wv,
    const float* __restrict__ vg, const float* __restrict__ vb,
    const float* __restrict__ vm, const float* __restrict__ vvar,
    float* __restrict__ qbuf,                             // [8,64,4096]
    float* __restrict__ kbuf,                             // [8,16,4096]
    float* __restrict__ vbuf)                             // [8,64,4096]
{
  __shared__ float sW[16 * 256];
  __shared__ float sS[16];
  __shared__ float sT[16];

  const int b  = blockIdx.z;
  const int rb = blockIdx.y;
  const int wave = threadIdx.x >> 5;
  const int lane = threadIdx.x & 31;
  const int j  = lane & 15;
  const bool hi = (lane & 16) != 0;
  const int n0 = blockIdx.x * 128 + wave * 16;

  // Stage the 16 weight rows for this row-block into LDS.
  for (int idx = threadIdx.x; idx < 16 * 256; idx += 256) {
    const int r = idx >> 8, c = idx & 255;
    const int grow = rb * 16 + r;
    float wv;
    if (grow < 64)       wv = Wq[grow * 256 + c];
    else if (grow < 80)  wv = Wk[(grow - 64) * 256 + c];
    else                 wv = Wv[(grow - 80) * 256 + c];
    sW[idx] = wv;
  }
  // Per-row folded BN scale/bias.
  if (threadIdx.x < 16) {
    const int grow = rb * 16 + threadIdx.x;
    float s, t;
    if (grow < 64) {
      const float inv = rsqrtf(qv[grow] + 1e-5f);
      s = qg[grow] * inv;
      t = qb[grow] - qm[grow] * s;
    } else if (grow < 80) {
      s = 1.0f; t = 0.0f;
    } else {
      const int r = grow - 80;
      const float inv = rsqrtf(vvar[r] + 1e-5f);
      s = vg[r] * inv;
      t = vb[r] - vm[r] * s;
    }
    sS[threadIdx.x] = s;
    sT[threadIdx.x] = t;
  }
  __syncthreads();

  const float* xb = x + (size_t)b * 256 * 4096;
  v8f acc = {};
  // Peeled K-loop: first 48 steps issue a prefetch 16 K-steps ahead
  // (global_prefetch_b8), last 16 steps are prefetch-free. Both inner
  // loops are branch-free.
  for (int kk = 0; kk < 48; ++kk) {
    const int kA0 = 4 * kk + (hi ? 2 : 0);
    const int kA1 = kA0 + 1;
    __builtin_prefetch(xb + (size_t)(4 * (kk + 16)) * 4096 + n0 + j, 0, 3);
    v2f a, bb;
    a.x  = sW[j * 256 + kA0];
    a.y  = sW[j * 256 + kA1];
    bb.x = xb[(size_t)kA0 * 4096 + n0 + j];
    bb.y = xb[(size_t)kA1 * 4096 + n0 + j];
    acc = WMMA_F32X4(a, bb, acc);
  }
  for (int kk = 48; kk < 64; ++kk) {
    const int kA0 = 4 * kk + (hi ? 2 : 0);
    const int kA1 = kA0 + 1;
    v2f a, bb;
    a.x  = sW[j * 256 + kA0];
    a.y  = sW[j * 256 + kA1];
    bb.x = xb[(size_t)kA0 * 4096 + n0 + j];
    bb.y = xb[(size_t)kA1 * 4096 + n0 + j];
    acc = WMMA_F32X4(a, bb, acc);
  }

  for (int r = 0; r < 8; ++r) {
    const int mr = r + (hi ? 8 : 0);
    const int grow = rb * 16 + mr;
    const float val = acc[r] * sS[mr] + sT[mr];
    const size_t col = (size_t)n0 + j;
    if (grow < 64)
      qbuf[((size_t)b * 64 + grow) * 4096 + col] = val;
    else if (grow < 80)
      kbuf[((size_t)b * 16 + (grow - 64)) * 4096 + col] = val;
    else
      vbuf[((size_t)b * 64 + (grow - 80)) * 4096 + col] = val;
  }
}

// ---------------------------------------------------------------------------
// Kernel 2: softmax over spatial positions (row length 4096).
// grid: 128 rows (b*16+k), block 256
// ---------------------------------------------------------------------------
__global__ __launch_bounds__(256) void softmax_kernel(
    const float* __restrict__ kbuf, float* __restrict__ smbuf)
{
  __shared__ float red[256];
  const int row = blockIdx.x;
  const float* in = kbuf + (size_t)row * 4096;
  float* outp = smbuf + (size_t)row * 4096;

  float mx = -INFINITY;
  for (int i = threadIdx.x; i < 4096; i += 256) mx = fmaxf(mx, in[i]);
  red[threadIdx.x] = mx;
  __syncthreads();
  for (int s = 128; s > 0; s >>= 1) {
    if (threadIdx.x < s) red[threadIdx.x] = fmaxf(red[threadIdx.x], red[threadIdx.x + s]);
    __syncthreads();
  }
  mx = red[0];
  __syncthreads();

  float sum = 0.0f;
  for (int i = threadIdx.x; i < 4096; i += 256) sum += __expf(in[i] - mx);
  red[threadIdx.x] = sum;
  __syncthreads();
  for (int s = 128; s > 0; s >>= 1) {
    if (threadIdx.x < s) red[threadIdx.x] += red[threadIdx.x + s];
    __syncthreads();
  }
  const float inv = 1.0f / red[0];
  for (int i = threadIdx.x; i < 4096; i += 256) outp[i] = __expf(in[i] - mx) * inv;
}

// ---------------------------------------------------------------------------
// Kernel 3: lam_c[b,k,v] = sum_n sm[b,k,n] * values[b,v,n]
// GEMM M=16(k) N=16(v-tile) K=4096, split-K over 8 waves + LDS reduce.
// grid: 32 (b*4 + vtile), block 256
// ---------------------------------------------------------------------------
__global__ __launch_bounds__(256) void lamc_kernel(
    const float* __restrict__ smbuf, const float* __restrict__ vbuf,
    float* __restrict__ lamc)
{
  __shared__ float sPart[8 * 256];
  const int b  = blockIdx.x >> 2;
  const int vt = blockIdx.x & 3;
  const int wave = threadIdx.x >> 5;
  const int lane = threadIdx.x & 31;
  const int j = lane & 15;
  const bool hi = (lane & 16) != 0;
  const int v0 = vt * 16;

  const float* smb = smbuf + (size_t)b * 16 * 4096;
  const float* vbb = vbuf + (size_t)b * 64 * 4096;

  v8f acc = {};
  for (int s = 0; s < 128; ++s) {
    const int n0 = wave * 512 + 4 * s + (hi ? 2 : 0);
    v2f a, bb;
    a.x  = smb[(size_t)j * 4096 + n0];
    a.y  = smb[(size_t)j * 4096 + n0 + 1];
    bb.x = vbb[(size_t)(v0 + j) * 4096 + n0];
    bb.y = vbb[(size_t)(v0 + j) * 4096 + n0 + 1];
    acc = WMMA_F32X4(a, bb, acc);
  }
  for (int r = 0; r < 8; ++r) {
    const int mr = r + (hi ? 8 : 0);
    sPart[wave * 256 + mr * 16 + j] = acc[r];
  }
  __syncthreads();
  float sum = 0.0f;
  for (int w = 0; w < 8; ++w) sum += sPart[w * 256 + threadIdx.x];
  lamc[((size_t)b * 16 + (threadIdx.x >> 4)) * 64 + v0 + (threadIdx.x & 15)] = sum;
}

// ---------------------------------------------------------------------------
// Kernel 4 (dominant FLOPs): im2col-WMMA 11x11 conv -> lam_p tile, fold in
// lam_c, contract with q over k, write final output.
// grid: (y=64, v=64, b=8), block 128 (4 waves; one 16-x tile per wave)
// ---------------------------------------------------------------------------
__global__ __launch_bounds__(128) void fused_kernel(
    const float* __restrict__ qbuf,     // [8,64,4096]  row = h*16+k
    const float* __restrict__ vbuf,     // [8,64,4096]
    const float* __restrict__ lamc,     // [8,16,64]
    const float* __restrict__ emb,      // [16,121]
    float* __restrict__ out)            // [8,256,64,64]
{
  __shared__ float sEmb[16 * 124];   // zero-padded taps 121..123
  __shared__ float sPatch[11 * 74];  // padded value patch: rows y-5..y+5, x -5..68
  __shared__ float sQ[64 * 64];      // q rows (h*16+k) x 64 x-positions
  __shared__ float sL[16 * 64];      // L[k][x] = lam_p + lam_c
  __shared__ float sLamC[16];

  const int y = blockIdx.x;
  const int v = blockIdx.y;
  const int b = blockIdx.z;
  const int wave = threadIdx.x >> 5;
  const int lane = threadIdx.x & 31;
  const int j = lane & 15;
  const bool hi = (lane & 16) != 0;
  const int x0 = wave * 16;

  // Async-stage q tile for this (b, y): 64 rows x 64 cols, 16B granules.
  // Each thread issues 8 GLOBAL_LOAD_ASYNC_TO_LDS_B128 (ASYNCcnt-tracked).
  {
    const float* qgbase = qbuf + (size_t)b * 64 * 4096 + (size_t)y * 64;
    const unsigned sQ_base = lds_byte_off(&sQ[0]);
    for (int i = 0; i < 8; ++i) {
      const int g = threadIdx.x + i * 128;       // float4 group id, 0..1023
      const int r = g >> 4;                      // q row (h*16+k)
      const int c = (g & 15) * 4;                // x column (16B aligned)
      async_g2lds_b128(sQ_base + (unsigned)g * 16u,
                       qgbase + (size_t)r * 4096 + c);
    }
  }

  // Stage embedding (zero-padded in K to 124).
  for (int idx = threadIdx.x; idx < 16 * 124; idx += 128) {
    const int k = idx / 124, t = idx - k * 124;
    sEmb[idx] = (t < 121) ? emb[k * 121 + t] : 0.0f;
  }
  // Stage padded value patch for this (b, v, y).
  const float* vrow = vbuf + ((size_t)b * 64 + v) * 4096;
  for (int idx = threadIdx.x; idx < 11 * 74; idx += 128) {
    const int r = idx / 74, c = idx - r * 74;
    const int yy = y + r - 5;
    const int xx = c - 5;
    float val = 0.0f;
    if (yy >= 0 && yy < 64 && xx >= 0 && xx < 64) val = vrow[yy * 64 + xx];
    sPatch[idx] = val;
  }
  if (threadIdx.x < 16)
    sLamC[threadIdx.x] = lamc[((size_t)b * 16 + threadIdx.x) * 64 + v];

  wait_async_zero();   // q tile resident in LDS
  __syncthreads();

  // im2col GEMM: D[16k x 16x] = emb[16 x 124] @ patch-cols[124 x 16]
  v8f acc = {};
  for (int s = 0; s < 31; ++s) {
    const int t0 = 4 * s + (hi ? 2 : 0);
    const int t1 = t0 + 1;
    const int t0c = (t0 < 121) ? t0 : 0;   // A is zero there; clamp B index
    const int t1c = (t1 < 121) ? t1 : 0;
    const int dy0 = t0c / 11, dx0 = t0c - dy0 * 11;
    const int dy1 = t1c / 11, dx1 = t1c - dy1 * 11;
    v2f a, bb;
    a.x  = sEmb[j * 124 + t0];
    a.y  = sEmb[j * 124 + t1];
    bb.x = sPatch[dy0 * 74 + x0 + j + dx0];
    bb.y = sPatch[dy1 * 74 + x0 + j + dx1];
    acc = WMMA_F32X4(a, bb, acc);
  }

  // L[k][x] = lam_p + lam_c
  for (int r = 0; r < 8; ++r) {
    const int k = r + (hi ? 8 : 0);
    sL[k * 64 + x0 + j] = acc[r] + sLamC[k];
  }
  __syncthreads();

  // out[b, h*64+v, y, x] = sum_k q[b,h,k,y*64+x] * L[k][x]
  const int xx = threadIdx.x & 63;
  const int hp = threadIdx.x >> 6;   // 0 or 1 -> handles 2 heads each
  for (int hh = 0; hh < 2; ++hh) {
    const int h = hp * 2 + hh;
    float a = 0.0f;
    for (int k = 0; k < 16; ++k)
      a = fmaf(sQ[(h * 16 + k) * 64 + xx], sL[k * 64 + xx], a);
    out[(((size_t)b * 256) + h * 64 + v) * 4096 + (size_t)y * 64 + xx] = a;
  }
}

// ---------------------------------------------------------------------------
extern "C" void kernel_launch(void* const* d_in, const int* in_sizes, int n_in,
                              void* d_out, int out_size, void* d_ws, size_t ws_size,
                              hipStream_t stream) {
  const float* x    = (const float*)d_in[0];
  const float* Wq   = (const float*)d_in[1];
  const float* qg   = (const float*)d_in[2];
  const float* qbta = (const float*)d_in[3];
  const float* qm   = (const float*)d_in[4];
  const float* qv   = (const float*)d_in[5];
  const float* Wk   = (const float*)d_in[6];
  const float* Wv   = (const float*)d_in[7];
  const float* vg   = (const float*)d_in[8];
  const float* vbta = (const float*)d_in[9];
  const float* vm   = (const float*)d_in[10];
  const float* vvar = (const float*)d_in[11];
  const float* emb  = (const float*)d_in[12];
  float* out = (float*)d_out;

  float* ws    = (float*)d_ws;
  float* qbuf  = ws;                                   // 8*64*4096
  float* kbuf  = qbuf  + (size_t)8 * 64 * 4096;        // 8*16*4096
  float* vbuf  = kbuf  + (size_t)8 * 16 * 4096;        // 8*64*4096
  float* smbuf = vbuf  + (size_t)8 * 64 * 4096;        // 8*16*4096
  float* lamc  = smbuf + (size_t)8 * 16 * 4096;        // 8*16*64

  proj_kernel<<<dim3(32, 9, 8), 256, 0, stream>>>(
      x, Wq, qg, qbta, qm, qv, Wk, Wv, vg, vbta, vm, vvar, qbuf, kbuf, vbuf);
  softmax_kernel<<<dim3(128), 256, 0, stream>>>(kbuf, smbuf);
  lamc_kernel<<<dim3(32), 256, 0, stream>>>(smbuf, vbuf, lamc);
  fused_kernel<<<dim3(64, 64, 8), 128, 0, stream>>>(qbuf, vbuf, lamc, emb, out);
}